// LlamaAttention_ModernKIVI_80049600463007
// MI455X (gfx1250) — compile-verified
//
#include <hip/hip_runtime.h>
#include <hip/hip_bf16.h>

// Problem constants (from reference)
#define B_      8
#define H_      32
#define HKV_    8
#define D_      128
#define HID_    4096
#define G_      32
#define SQK_    8128
#define KFULL_  63
#define SQV_    8064
#define VFULL_  127
#define STOT_   8192                 // SQK_ + KFULL_ + 1
#define NGRPK_  (SQK_ / G_)          // 254 scale groups along S for K
#define NQKV_   (HID_ + 2 * HKV_ * D_) // 6144 concatenated qkv columns
#define KSPLIT_ 8
#define KLEN_   (HID_ / KSPLIT_)     // 512
#define SCALE_  0.08838834764831845f // 1/sqrt(128)
#define VCHUNK_ 1008                 // 8064 / 8
#define NVCH_   8

typedef float v2f __attribute__((ext_vector_type(2)));
typedef float v8f __attribute__((ext_vector_type(8)));

// ---------------------------------------------------------------------------
// M=8 (padded to 16) GEMM partial: C_part[ks][8][Ntot] = A[8xKLEN] * W tile.
// Uses V_WMMA_F32_16X16X4_F32 (full fp32 precision on the matrix pipe).
// A frag (16x4): lanes 0-15 hold K=0(VGPR0),1(VGPR1); lanes 16-31 hold K=2,3.
// B frag (4x16): lane-in-group = N; lanes 0-15 rows K=0,1; lanes 16-31 K=2,3.
// C (16x16): VGPR r, lanes 0-15 -> (M=r, N=lane).
//
// NPART is a template constant so the W row stride is a literal (no v_mul_u64
// in the loop); the A tile is zero-padded to 16 rows so the A fragment is an
// unconditional 8B-aligned ds_load_b64; the loop is software-pipelined depth 2
// so global B-fragment loads overlap the previous WMMA.
// ---------------------------------------------------------------------------
template <int NPART>
__global__ __launch_bounds__(256)
void gemm8_wmma_part(const float* __restrict__ A, const float* __restrict__ W,
                     int coloff, int Ntot, float* __restrict__ part) {
    __shared__ float hid[16][KLEN_];
    const int tid = threadIdx.x;
    const int ks  = blockIdx.y;
    const int k0  = ks * KLEN_;
    for (int i = tid; i < 16 * KLEN_; i += 256) {
        int r = i / KLEN_, kk = i % KLEN_;
        hid[r][kk] = (r < 8) ? A[(size_t)r * HID_ + k0 + kk] : 0.f;
    }
    __syncthreads();

    const int wave = tid >> 5;
    const int lane = tid & 31;
    const int ln   = lane & 15;   // N within tile (B/C), M row (A)
    const int lg   = lane >> 4;   // K-half selector
    const int c0   = blockIdx.x * 128 + wave * 16;

    // Per-lane W base: row (k0 + 2*lg), column (c0 + ln).
    const float* base = W + ((size_t)k0 + 2 * lg) * NPART + c0 + ln;

    v8f acc = {0.f, 0.f, 0.f, 0.f, 0.f, 0.f, 0.f, 0.f};
    v2f a_cur, b_cur;
    a_cur = *(const v2f*)&hid[ln][2 * lg];
    b_cur.x = base[0];
    b_cur.y = base[NPART];

#pragma unroll 4
    for (int kk = 0; kk < KLEN_; kk += 4) {
        v2f a_nxt = a_cur, b_nxt = b_cur;
        if (kk + 4 < KLEN_) {  // uniform branch: prefetch next fragments
            a_nxt = *(const v2f*)&hid[ln][kk + 4 + 2 * lg];
            const float* bp = base + (size_t)(kk + 4) * NPART;
            b_nxt.x = bp[0];
            b_nxt.y = bp[NPART];
        }
        acc = __builtin_amdgcn_wmma_f32_16x16x4_f32(
            false, a_cur, false, b_cur, (short)0, acc, false, false);
        a_cur = a_nxt;
        b_cur = b_nxt;
    }

    if (lg == 0) {  // lanes 0-15 hold rows 0..7 (rows 8..15 are zero padding)
        float* op = part + (size_t)ks * 8 * Ntot + coloff + c0 + ln;
#pragma unroll
        for (int r = 0; r < 8; ++r) op[(size_t)r * Ntot] = acc[r];
    }
}

// ---------------------------------------------------------------------------
// Reduce split-K partials of [Wq|Wk|Wv] and apply RoPE; scatter to q/k/v ws.
// ---------------------------------------------------------------------------
__global__ __launch_bounds__(256)
void rope_scatter(const float* __restrict__ part, const float* __restrict__ cosp,
                  const float* __restrict__ sinp, float* __restrict__ q_ws,
                  float* __restrict__ k_ws, float* __restrict__ v_ws) {
    int idx = blockIdx.x * 256 + threadIdx.x;
    if (idx >= B_ * NQKV_) return;
    int b = idx / NQKV_;
    int c = idx % NQKV_;
    int d = c & 127;
    int cp = (d < 64) ? (c + 64) : (c - 64);   // rotate_half partner column
    float x = 0.f, xr = 0.f;
#pragma unroll
    for (int ks = 0; ks < KSPLIT_; ++ks) {
        const float* p = part + ((size_t)ks * B_ + b) * NQKV_;
        x  += p[c];
        xr += p[cp];
    }
    if (c >= HID_ + HKV_ * D_) {                       // V: no RoPE
        int kvh = (c - HID_ - HKV_ * D_) >> 7;
        v_ws[((size_t)b * HKV_ + kvh) * D_ + d] = x;
    } else {
        float cs  = cosp[b * D_ + d];
        float sn  = sinp[b * D_ + d];
        float rot = (d < 64) ? -xr : xr;
        float val = x * cs + rot * sn;
        if (c < HID_) {
            int h = c >> 7;
            q_ws[((size_t)b * H_ + h) * D_ + d] = val;
        } else {
            int kvh = (c - HID_) >> 7;
            k_ws[((size_t)b * HKV_ + kvh) * D_ + d] = val;
        }
    }
}

// ---------------------------------------------------------------------------
// Scores vs 2-bit quantized K (transposed layout: k_quant_t[b,kvh,d,s]).
// One block = (b,kvh, 256 s positions); the 4 q-heads of the group share the
// K stream (read once). Per d: lanes read 32 consecutive s (coalesced);
// scale/min addresses are wave-uniform (coalesce to a single request).
// ---------------------------------------------------------------------------
__global__ __launch_bounds__(256)
void scores_quant(const float* __restrict__ q_ws, const int* __restrict__ kq,
                  const float* __restrict__ ksc, const float* __restrict__ kmn,
                  float* __restrict__ scores) {
    __shared__ float ql[4][D_];
    const int tid = threadIdx.x;
    const int y = blockIdx.y;          // b*HKV + kvh
    const int b = y >> 3, kvh = y & 7;
    for (int i = tid; i < 4 * D_; i += 256)
        ql[i >> 7][i & 127] =
            q_ws[(((size_t)b * H_ + (kvh << 2) + (i >> 7)) << 7) + (i & 127)];
    __syncthreads();

    const int s = blockIdx.x * 256 + tid;
    if (s >= SQK_) return;
    const int g = s >> 5;                       // 32 positions share scale/min
    const size_t kb = (size_t)y * D_ * SQK_;
    const size_t sb = (size_t)y * D_ * NGRPK_;
    float a0 = 0.f, a1 = 0.f, a2 = 0.f, a3 = 0.f;
#pragma unroll 4
    for (int d = 0; d < D_; ++d) {
        float kval = fmaf((float)kq[kb + (size_t)d * SQK_ + s],
                          ksc[sb + d * NGRPK_ + g], kmn[sb + d * NGRPK_ + g]);
        a0 = fmaf(ql[0][d], kval, a0);
        a1 = fmaf(ql[1][d], kval, a1);
        a2 = fmaf(ql[2][d], kval, a2);
        a3 = fmaf(ql[3][d], kval, a3);
    }
    float* sp = scores + ((size_t)b * H_ + (kvh << 2)) * STOT_ + s;
    sp[0]         = a0 * SCALE_;
    sp[STOT_]     = a1 * SCALE_;
    sp[2 * STOT_] = a2 * SCALE_;
    sp[3 * STOT_] = a3 * SCALE_;
}

// ---------------------------------------------------------------------------
// Scores vs full-precision K (63 cached rows + 1 freshly computed row).
// ---------------------------------------------------------------------------
__global__ __launch_bounds__(256)
void scores_full(const float* __restrict__ q_ws, const float* __restrict__ kfull,
                 const float* __restrict__ k_ws, float* __restrict__ scores) {
    __shared__ float ql[4][D_];
    const int tid = threadIdx.x;
    const int y = blockIdx.x; const int b = y >> 3, kvh = y & 7;
    for (int i = tid; i < 4 * D_; i += 256)
        ql[i >> 7][i & 127] =
            q_ws[(((size_t)b * H_ + (kvh << 2) + (i >> 7)) << 7) + (i & 127)];
    __syncthreads();
    const int h4 = tid >> 6, j = tid & 63;
    const float* krow = (j < KFULL_) ? (kfull + (((size_t)y * KFULL_ + j) << 7))
                                     : (k_ws + ((size_t)y << 7));
    float a = 0.f;
#pragma unroll 4
    for (int d = 0; d < D_; ++d) a = fmaf(ql[h4][d], krow[d], a);
    scores[((size_t)b * H_ + (kvh << 2) + h4) * STOT_ + SQK_ + j] = a * SCALE_;
}

// ---------------------------------------------------------------------------
// In-place softmax over 8192 positions, one block per (b,h).
// ---------------------------------------------------------------------------
__global__ __launch_bounds__(256)
void softmax_k(float* __restrict__ scores) {
    __shared__ float red[256];
    float* row = scores + (size_t)blockIdx.x * STOT_;
    const int tid = threadIdx.x;
    float m = -1e30f;
    for (int i = tid; i < STOT_; i += 256) m = fmaxf(m, row[i]);
    red[tid] = m; __syncthreads();
    for (int o = 128; o > 0; o >>= 1) {
        if (tid < o) red[tid] = fmaxf(red[tid], red[tid + o]);
        __syncthreads();
    }
    m = red[0]; __syncthreads();
    float sum = 0.f;
    for (int i = tid; i < STOT_; i += 256) {
        float e = __expf(row[i] - m);
        row[i] = e;
        sum += e;
    }
    red[tid] = sum; __syncthreads();
    for (int o = 128; o > 0; o >>= 1) {
        if (tid < o) red[tid] += red[tid + o];
        __syncthreads();
    }
    float inv = 1.f / red[0];
    for (int i = tid; i < STOT_; i += 256) row[i] *= inv;
}

// ---------------------------------------------------------------------------
// A·V over quantized V (v_quant[b,kvh,s,d], scale groups along d).
// One block = (b,kvh, 1008 s rows). Probs staged in LDS; vq reads coalesced.
// Partial out -> out_part[chunk][b][h][d] (deterministic, no atomics).
// ---------------------------------------------------------------------------
__global__ __launch_bounds__(256)
void av_quant(const float* __restrict__ scores, const int* __restrict__ vq,
              const float* __restrict__ vsc, const float* __restrict__ vmn,
              float* __restrict__ out_part) {
    __shared__ float pl[4][VCHUNK_];
    __shared__ float red[2][4][D_];
    const int tid = threadIdx.x;
    const int y = blockIdx.y, b = y >> 3, kvh = y & 7;
    const int s0 = blockIdx.x * VCHUNK_;
    for (int i = tid; i < 4 * VCHUNK_; i += 256) {
        int h = i / VCHUNK_, s = i % VCHUNK_;
        pl[h][s] = scores[((size_t)b * H_ + (kvh << 2) + h) * STOT_ + s0 + s];
    }
    __syncthreads();

    const int d = tid & 127, sh = tid >> 7;
    const size_t vb = (size_t)y * SQV_ * D_;
    const size_t sb = (size_t)y * SQV_ * 4;
    float a0 = 0.f, a1 = 0.f, a2 = 0.f, a3 = 0.f;
    for (int s = sh; s < VCHUNK_; s += 2) {
        const int srow = s0 + s;
        float val = fmaf((float)vq[vb + (size_t)srow * D_ + d],
                         vsc[sb + srow * 4 + (d >> 5)],
                         vmn[sb + srow * 4 + (d >> 5)]);
        a0 = fmaf(pl[0][s], val, a0);
        a1 = fmaf(pl[1][s], val, a1);
        a2 = fmaf(pl[2][s], val, a2);
        a3 = fmaf(pl[3][s], val, a3);
    }
    red[sh][0][d] = a0; red[sh][1][d] = a1; red[sh][2][d] = a2; red[sh][3][d] = a3;
    __syncthreads();
    for (int i = tid; i < 512; i += 256) {
        int h = i >> 7, dd = i & 127;
        out_part[(((size_t)blockIdx.x * B_ + b) * H_ + (kvh << 2) + h) * D_ + dd] =
            red[0][h][dd] + red[1][h][dd];
    }
}

// ---------------------------------------------------------------------------
// A·V over full-precision V (127 cached rows + 1 new row); chunk slot 8.
// Probabilities p[8064 + j] pair with v_full_cat[j], j in [0,128).
// ---------------------------------------------------------------------------
__global__ __launch_bounds__(256)
void av_full(const float* __restrict__ scores, const float* __restrict__ vfull,
             const float* __restrict__ v_ws, float* __restrict__ out_part) {
    __shared__ float pl[4][128];
    __shared__ float red[2][4][D_];
    const int tid = threadIdx.x;
    const int y = blockIdx.x, b = y >> 3, kvh = y & 7;
    for (int i = tid; i < 512; i += 256) {
        int h = i >> 7, j = i & 127;
        pl[h][j] = scores[((size_t)b * H_ + (kvh << 2) + h) * STOT_ + SQV_ + j];
    }
    __syncthreads();
    const int d = tid & 127, jh = tid >> 7;
    float a0 = 0.f, a1 = 0.f, a2 = 0.f, a3 = 0.f;
    for (int j = jh; j < 128; j += 2) {
        const float* vrow = (j < VFULL_) ? (vfull + (((size_t)y * VFULL_ + j) << 7))
                                         : (v_ws + ((size_t)y << 7));
        float val = vrow[d];
        a0 = fmaf(pl[0][j], val, a0);
        a1 = fmaf(pl[1][j], val, a1);
        a2 = fmaf(pl[2][j], val, a2);
        a3 = fmaf(pl[3][j], val, a3);
    }
    red[jh][0][d] = a0; red[jh][1][d] = a1; red[jh][2][d] = a2; red[jh][3][d] = a3;
    __syncthreads();
    for (int i = tid; i < 512; i += 256) {
        int h = i >> 7, dd = i & 127;
        out_part[(((size_t)NVCH_ * B_ + b) * H_ + (kvh << 2) + h) * D_ + dd] =
            red[0][h][dd] + red[1][h][dd];
    }
}

// ---------------------------------------------------------------------------
// Reduce the 9 A·V partial slots into attn_out[b][h][d] == A rows for Wo.
// ---------------------------------------------------------------------------
__global__ __launch_bounds__(256)
void av_reduce(const float* __restrict__ out_part, float* __restrict__ attn_out) {
    int idx = blockIdx.x * 256 + threadIdx.x;
    float s = 0.f;
#pragma unroll
    for (int c = 0; c < 9; ++c) s += out_part[(size_t)c * (B_ * H_ * D_) + idx];
    attn_out[idx] = s;
}

// ---------------------------------------------------------------------------
// Reduce split-K partials of the output projection into d_out.
// ---------------------------------------------------------------------------
__global__ __launch_bounds__(256)
void part_reduce8(const float* __restrict__ part, float* __restrict__ out, int n) {
    int idx = blockIdx.x * 256 + threadIdx.x;
    if (idx >= n) return;
    float s = 0.f;
#pragma unroll
    for (int c = 0; c < KSPLIT_; ++c) s += part[(size_t)c * n + idx];
    out[idx] = s;
}

extern "C" void kernel_launch(void* const* d_in, const int* in_sizes, int n_in,
                              void* d_out, int out_size, void* d_ws, size_t ws_size,
                              hipStream_t stream) {
    const float* hidden = (const float*)d_in[0];
    const float* Wq     = (const float*)d_in[1];
    const float* Wk     = (const float*)d_in[2];
    const float* Wv     = (const float*)d_in[3];
    const float* Wo     = (const float*)d_in[4];
    const float* cosp   = (const float*)d_in[5];
    const float* sinp   = (const float*)d_in[6];
    const int*   kq     = (const int*)d_in[7];
    const float* ksc    = (const float*)d_in[8];
    const float* kmn    = (const float*)d_in[9];
    const float* kfull  = (const float*)d_in[10];
    const int*   vq     = (const int*)d_in[11];
    const float* vsc    = (const float*)d_in[12];
    const float* vmn    = (const float*)d_in[13];
    const float* vfull  = (const float*)d_in[14];
    float* out = (float*)d_out;

    // Workspace carve-up (~12.6 MB of f32)
    float* ws         = (float*)d_ws;
    float* q_ws       = ws;                                       // B*H*D
    float* k_ws       = q_ws + B_ * H_ * D_;                      // B*HKV*D
    float* v_ws       = k_ws + B_ * HKV_ * D_;                    // B*HKV*D
    float* scores     = v_ws + B_ * HKV_ * D_;                    // B*H*STOT
    float* xqkv_part  = scores + (size_t)B_ * H_ * STOT_;         // KSPLIT*B*NQKV
    float* attn_out   = xqkv_part + (size_t)KSPLIT_ * B_ * NQKV_; // B*H*D
    float* out_part   = attn_out + B_ * H_ * D_;                  // 9*B*H*D
    float* oproj_part = out_part + 9 * B_ * H_ * D_;              // KSPLIT*B*HID

    // 1) QKV projections (WMMA f32 16x16x4, split-K partials, every slot written)
    gemm8_wmma_part<HID_><<<dim3(HID_ / 128, KSPLIT_), 256, 0, stream>>>(
        hidden, Wq, 0, NQKV_, xqkv_part);
    gemm8_wmma_part<HKV_ * D_><<<dim3(HKV_ * D_ / 128, KSPLIT_), 256, 0, stream>>>(
        hidden, Wk, HID_, NQKV_, xqkv_part);
    gemm8_wmma_part<HKV_ * D_><<<dim3(HKV_ * D_ / 128, KSPLIT_), 256, 0, stream>>>(
        hidden, Wv, HID_ + HKV_ * D_, NQKV_, xqkv_part);
    // 2) reduce + RoPE + scatter
    rope_scatter<<<(B_ * NQKV_ + 255) / 256, 256, 0, stream>>>(
        xqkv_part, cosp, sinp, q_ws, k_ws, v_ws);
    // 3) attention scores
    scores_quant<<<dim3((SQK_ + 255) / 256, B_ * HKV_), 256, 0, stream>>>(
        q_ws, kq, ksc, kmn, scores);
    scores_full<<<B_ * HKV_, 256, 0, stream>>>(q_ws, kfull, k_ws, scores);
    // 4) softmax
    softmax_k<<<B_ * H_, 256, 0, stream>>>(scores);
    // 5) A·V (split-S partials + reduction; deterministic, no fp atomics)
    av_quant<<<dim3(NVCH_, B_ * HKV_), 256, 0, stream>>>(scores, vq, vsc, vmn, out_part);
    av_full<<<B_ * HKV_, 256, 0, stream>>>(scores, vfull, v_ws, out_part);
    av_reduce<<<(B_ * H_ * D_) / 256, 256, 0, stream>>>(out_part, attn_out);
    // 6) output projection (WMMA) + final reduce
    gemm8_wmma_part<HID_><<<dim3(HID_ / 128, KSPLIT_), 256, 0, stream>>>(
        attn_out, Wo, 0, HID_, oproj_part);
    part_reduce8<<<(B_ * HID_) / 256, 256, 0, stream>>>(oproj_part, out, B_ * HID_);
}